// CoulombPotential_79860621902321
// MI455X (gfx1250) — compile-verified
//
#include <hip/hip_runtime.h>

// ---------------------------------------------------------------------------
// CoulombPotential on MI455X (gfx1250)
//   - streams pair_idx_i / pair_idx_j / d_ij via async global->LDS DMA
//     (double-buffered, ASYNCcnt-tracked) to overlap with random gathers
//   - per-workgroup LDS histogram (ds_add_f32) over S system bins
//   - global_atomic_add_f32 flush with KE folded in
// ---------------------------------------------------------------------------

#define BLOCK   256
#define CHUNK   1024          // pairs per chunk = BLOCK * 4 (one b128 per lane/stream)
#define MAXBINS 2048
#define KE_F    138.96f

#define AS1 __attribute__((address_space(1)))
#define AS3 __attribute__((address_space(3)))

typedef int v4i __attribute__((vector_size(16)));

__device__ __forceinline__ void async_copy_b128(const void* gsrc, void* ldst) {
#if __has_builtin(__builtin_amdgcn_global_load_async_to_lds_b128)
  __builtin_amdgcn_global_load_async_to_lds_b128(
      (AS1 v4i*)gsrc, (AS3 v4i*)ldst, 0, 0);
#else
  unsigned laddr = (unsigned)(__SIZE_TYPE__)(AS3 void*)ldst;
  unsigned long long ga = (unsigned long long)gsrc;
  asm volatile("global_load_async_to_lds_b128 %0, %1, off"
               :: "v"(laddr), "v"(ga) : "memory");
#endif
}

template <int N>
__device__ __forceinline__ void wait_asynccnt() {
#if __has_builtin(__builtin_amdgcn_s_wait_asynccnt)
  __builtin_amdgcn_s_wait_asynccnt(N);
#else
  asm volatile("s_wait_asynccnt %0" :: "i"(N) : "memory");
#endif
}

__device__ __forceinline__ float pair_energy(float d, float qi, float qj) {
  // phi(u) = 1 - 10u^3 + 15u^4 - 6u^5,  u = 2d (cutoff = 1), zero for u >= 1
  float u   = 2.0f * d;
  float phi = (u < 1.0f)
      ? fmaf(u * u * u, fmaf(u, fmaf(-6.0f, u, 15.0f), -10.0f), 1.0f)
      : 0.0f;
  float chi = phi * __builtin_amdgcn_rsqf(fmaf(d, d, 1.0f))
            + (1.0f - phi) * __builtin_amdgcn_rcpf(d);
  return qi * qj * chi;
}

__global__ void __launch_bounds__(BLOCK) coulomb_zero(float* out, int S) {
  int i = blockIdx.x * BLOCK + threadIdx.x;
  if (i < S) out[i] = 0.0f;
}

__global__ void __launch_bounds__(BLOCK) coulomb_main(
    const int*   __restrict__ pairI,
    const int*   __restrict__ pairJ,
    const float* __restrict__ dij,
    const float* __restrict__ q,
    const int*   __restrict__ sys,
    float*       __restrict__ out,
    int P, int S, int nFull)
{
  __shared__ int   sI[2][CHUNK];
  __shared__ int   sJ[2][CHUNK];
  __shared__ float sD[2][CHUNK];
  __shared__ float bins[MAXBINS];

  const int  t      = threadIdx.x;
  const bool useLds = (S <= MAXBINS);

  if (useLds) {
    for (int s = t; s < S; s += BLOCK) bins[s] = 0.0f;
  }
  __syncthreads();

  // ---------------- software-pipelined chunk loop (full chunks) ------------
  unsigned       c   = blockIdx.x;
  const unsigned g   = gridDim.x;
  int            buf = 0;
  bool           have = ((int)c < nFull);

  if (have) {
    size_t base = (size_t)c * CHUNK + 4u * (unsigned)t;
    async_copy_b128(pairI + base, &sI[0][4 * t]);
    async_copy_b128(pairJ + base, &sJ[0][4 * t]);
    async_copy_b128(dij   + base, &sD[0][4 * t]);
  }

  while (have) {
    unsigned cn       = c + g;
    bool     haveNext = ((int)cn < nFull);
    if (haveNext) {
      size_t base = (size_t)cn * CHUNK + 4u * (unsigned)t;
      int    nb   = buf ^ 1;
      async_copy_b128(pairI + base, &sI[nb][4 * t]);
      async_copy_b128(pairJ + base, &sJ[nb][4 * t]);
      async_copy_b128(dij   + base, &sD[nb][4 * t]);
      wait_asynccnt<3>();   // current chunk's 3 copies done; next 3 in flight
    } else {
      wait_asynccnt<0>();
    }
    __syncthreads();

#pragma unroll
    for (int m = 0; m < 4; ++m) {
      int li = t + m * BLOCK;
      int i  = sI[buf][li];
      int j  = sJ[buf][li];
      if (i < j) {                       // EXEC-masks the gathers too
        float d  = sD[buf][li];
        float qi = q[i];
        float qj = q[j];
        int   sg = sys[i];
        float e  = pair_energy(d, qi, qj);
        if (useLds) atomicAdd(&bins[sg], e);           // ds_add_f32
        else        unsafeAtomicAdd(&out[sg], e * KE_F);
      }
    }
    __syncthreads();   // all lanes done reading buf before it is re-filled

    c = cn;
    buf ^= 1;
    have = haveNext;
  }

  // ---------------- tail (P not a multiple of CHUNK), block 0 only ---------
  if (blockIdx.x == 0) {
    for (int p = nFull * CHUNK + t; p < P; p += BLOCK) {
      int i = pairI[p];
      int j = pairJ[p];
      if (i < j) {
        float e  = pair_energy(dij[p], q[i], q[j]);
        int   sg = sys[i];
        if (useLds) atomicAdd(&bins[sg], e);
        else        unsafeAtomicAdd(&out[sg], e * KE_F);
      }
    }
  }

  // ---------------- flush LDS histogram to global ----------------
  if (useLds) {
    __syncthreads();
    for (int s = t; s < S; s += BLOCK) {
      unsafeAtomicAdd(&out[s], bins[s] * KE_F);        // global_atomic_add_f32
    }
  }
}

extern "C" void kernel_launch(void* const* d_in, const int* in_sizes, int n_in,
                              void* d_out, int out_size, void* d_ws, size_t ws_size,
                              hipStream_t stream) {
  (void)n_in; (void)d_ws; (void)ws_size;

  const int*   pairs = (const int*)  d_in[0];   // (2, P) int32
  const float* dij   = (const float*)d_in[1];   // (P, 1) f32
  const float* q     = (const float*)d_in[2];   // (N, 1) f32
  const int*   sys   = (const int*)  d_in[3];   // (N,)   int32
  float*       out   = (float*)d_out;           // (S, 1) f32

  const int P = in_sizes[0] / 2;
  const int S = out_size;

  const int* pairI = pairs;
  const int* pairJ = pairs + P;

  int nFull = P / CHUNK;
  int grid  = nFull < 1024 ? (nFull > 0 ? nFull : 1) : 1024;

  coulomb_zero<<<(S + BLOCK - 1) / BLOCK, BLOCK, 0, stream>>>(out, S);
  coulomb_main<<<grid, BLOCK, 0, stream>>>(pairI, pairJ, dij, q, sys, out,
                                           P, S, nFull);
}